// GraphConvolution_62070867362377
// MI455X (gfx1250) — compile-verified
//
#include <hip/hip_runtime.h>
#include <hip/hip_bf16.h>

#define N_NODES 50000
#define N_EDGES 600000
#define D 128

typedef __attribute__((ext_vector_type(2))) float v2f;
typedef __attribute__((ext_vector_type(8))) float v8f;

// Guaranteed-hardware f32 atomic add (CDNA5 GLOBAL_ATOMIC_ADD_F32, opcode 86,
// non-returning form -> tracked on STOREcnt; s_endpgm implicitly waits idle).
// Avoids any chance of a CAS-loop expansion from the HIP header path.
__device__ __forceinline__ void atomic_add_f32_hw(float* addr, float val) {
  asm volatile("global_atomic_add_f32 %0, %1, off"
               :
               : "v"((unsigned long long)(uintptr_t)addr), "v"(val)
               : "memory");
}

// ---------------------------------------------------------------------------
// Phase 3: COO SpMM scatter. One wave32 per edge; each lane owns a float4
// column chunk (32 lanes x 4 = 128 cols). Gather pre[col] (coalesced 512B),
// scale by edge value, global_atomic_add_f32 into out[row]. The working set
// (pre0/pre1 + out ~ 77 MB) is L2-resident on MI455X (192 MB L2), so both the
// gather and the atomic scatter run at L2 rate, not HBM rate.
// ---------------------------------------------------------------------------
__global__ __launch_bounds__(256) void spmm_edge_atomic(
    const int* __restrict__ rows,
    const int* __restrict__ cols,
    const float* __restrict__ vals,
    const float* __restrict__ pre,
    float* __restrict__ out,
    int n_edges) {
  const int gid  = blockIdx.x * blockDim.x + threadIdx.x;
  const int e    = gid >> 5;
  const int lane = gid & 31;
  if (e >= n_edges) return;

  const int   r = rows[e];
  const int   c = cols[e];
  const float v = vals[e];

  const float4 p = *(const float4*)(pre + (size_t)c * D + lane * 4);
  float* o = out + (size_t)r * D + lane * 4;

  atomic_add_f32_hw(o + 0, v * p.x);
  atomic_add_f32_hw(o + 1, v * p.y);
  atomic_add_f32_hw(o + 2, v * p.z);
  atomic_add_f32_hw(o + 3, v * p.w);
}

// ---------------------------------------------------------------------------
// Phase 2: out[r][c] = b[c]  (bias pre-seeded so SpMM atomics accumulate on top)
// ---------------------------------------------------------------------------
__global__ __launch_bounds__(256) void init_bias(
    float* __restrict__ out, const float* __restrict__ b) {
  const int i = blockIdx.x * blockDim.x + threadIdx.x;
  if (i < N_NODES * D) out[i] = b[i & (D - 1)];
}

// ---------------------------------------------------------------------------
// Phase 4: out = relu(out)
// ---------------------------------------------------------------------------
__global__ __launch_bounds__(256) void relu_inplace(float* __restrict__ out) {
  const int i = blockIdx.x * blockDim.x + threadIdx.x;
  if (i < N_NODES * D) out[i] = fmaxf(out[i], 0.0f);
}

// ---------------------------------------------------------------------------
// Phase 1: pre0 = x @ W0, pre1 = x @ W1 using V_WMMA_F32_16X16X4_F32.
// One wave per 16x16 output tile (blockIdx.x = M tile, wave-in-block = N tile).
// A fragment (16x4 f32): lanes 0-15 hold M=0..15 / K={kt,kt+1} in v0,v1;
//                        lanes 16-31 hold M=0..15 / K={kt+2,kt+3}.
// B fragment (4x16 f32): mirrored with N instead of M.
// C/D (16x16 f32): VGPR v -> M=v (lanes 0-15), M=v+8 (lanes 16-31); N = lane&15.
// One A fragment feeds two WMMAs (W0 and W1 accumulators) for x-reuse.
// ---------------------------------------------------------------------------
__global__ __launch_bounds__(256) void gemm_dual_wmma(
    const float* __restrict__ x,
    const float* __restrict__ W0,
    const float* __restrict__ W1,
    float* __restrict__ pre0,
    float* __restrict__ pre1) {
  const int lane = threadIdx.x & 31;
  const int wave = threadIdx.x >> 5;          // 0..7 -> N tile
  const int m0   = blockIdx.x * 16;           // 3125 M tiles (50000/16)
  const int n0   = wave * 16;                 // 8 N tiles (128/16)
  const int half = lane >> 4;                 // 0 or 1 (K sub-pair selector)
  const int l    = lane & 15;

  const float* xrow = x + (size_t)(m0 + l) * D;   // A-matrix row for this lane
  const int n = n0 + l;                           // B/C column for this lane

  v8f c0 = {};
  v8f c1 = {};

  #pragma unroll 4
  for (int kt = 0; kt < D; kt += 4) {
    const int kb = kt + half * 2;
    v2f a = *(const v2f*)(xrow + kb);           // K = kb, kb+1 for row m0+l

    v2f b0, b1;
    b0.x = W0[(size_t)kb * D + n];
    b0.y = W0[(size_t)(kb + 1) * D + n];
    b1.x = W1[(size_t)kb * D + n];
    b1.y = W1[(size_t)(kb + 1) * D + n];

    c0 = __builtin_amdgcn_wmma_f32_16x16x4_f32(false, a, false, b0,
                                               (short)0, c0, false, false);
    c1 = __builtin_amdgcn_wmma_f32_16x16x4_f32(false, a, false, b1,
                                               (short)0, c1, false, false);
  }

  #pragma unroll
  for (int v = 0; v < 8; ++v) {
    const int row = m0 + v + half * 8;
    pre0[(size_t)row * D + n] = c0[v];
    pre1[(size_t)row * D + n] = c1[v];
  }
}

extern "C" void kernel_launch(void* const* d_in, const int* in_sizes, int n_in,
                              void* d_out, int out_size, void* d_ws, size_t ws_size,
                              hipStream_t stream) {
  const float* x     = (const float*)d_in[0];
  const float* W0    = (const float*)d_in[1];
  const float* W1    = (const float*)d_in[2];
  const float* b     = (const float*)d_in[3];
  const int*   rows0 = (const int*)d_in[4];
  const int*   cols0 = (const int*)d_in[5];
  const float* vals0 = (const float*)d_in[6];
  const int*   rows1 = (const int*)d_in[7];
  const int*   cols1 = (const int*)d_in[8];
  const float* vals1 = (const float*)d_in[9];
  float* out = (float*)d_out;

  float* pre0 = (float*)d_ws;                       // 50000*128 f32 = 25.6 MB
  float* pre1 = pre0 + (size_t)N_NODES * D;         // another 25.6 MB

  // Phase 1: dual GEMM via WMMA (3125 M-tiles x 8 N-tile waves per block)
  gemm_dual_wmma<<<N_NODES / 16, 256, 0, stream>>>(x, W0, W1, pre0, pre1);

  // Phase 2: seed output with bias
  init_bias<<<(N_NODES * D + 255) / 256, 256, 0, stream>>>(out, b);

  // Phase 3: edge-parallel scatter SpMM, one wave per edge
  const int spmm_blocks = (N_EDGES * 32 + 255) / 256;
  spmm_edge_atomic<<<spmm_blocks, 256, 0, stream>>>(rows0, cols0, vals0, pre0, out, N_EDGES);
  spmm_edge_atomic<<<spmm_blocks, 256, 0, stream>>>(rows1, cols1, vals1, pre1, out, N_EDGES);

  // Phase 4: relu
  relu_inplace<<<(N_NODES * D + 255) / 256, 256, 0, stream>>>(out);
}